// NodeNetwork_62818191671973
// MI455X (gfx1250) — compile-verified
//
#include <hip/hip_runtime.h>

#define Bv 4
#define Nv 2048
#define Ev 8192
#define Fv 16
#define Uv 64

typedef float v2f  __attribute__((ext_vector_type(2)));
typedef float v8f  __attribute__((ext_vector_type(8)));
typedef float vf4  __attribute__((ext_vector_type(4)));

// ---------------- degree kernels ----------------

__global__ void zero_ws(float* __restrict__ p, int n) {
    int i = blockIdx.x * blockDim.x + threadIdx.x;
    if (i < n) p[i] = 0.0f;
}

// Process one float4 of the one-hot matrix at flat float4-index i.
__device__ __forceinline__ void deg_update(vf4 v, int i, const float* __restrict__ ew,
                                           float* __restrict__ deg) {
    if (v.x != 0.0f || v.y != 0.0f || v.z != 0.0f || v.w != 0.0f) {
        int flat = i << 2;
        int n    = flat & (Nv - 1);   // N = 2048
        int be   = flat >> 11;        // / N
        int b    = be >> 13;          // / E (E = 8192)
        float w  = ew[be];
        float* d = deg + b * Nv;
        if (v.x != 0.0f) atomicAdd(d + n + 0, v.x * w);
        if (v.y != 0.0f) atomicAdd(d + n + 1, v.y * w);
        if (v.z != 0.0f) atomicAdd(d + n + 2, v.z * w);
        if (v.w != 0.0f) atomicAdd(d + n + 3, v.w * w);
    }
}

// Stream one dense one-hot matrix R [B,E,N] (268MB) and accumulate weighted
// degrees deg[b*N+n] += R[b,e,n] * ew[b,e]. Nonzeros are 1 in 2048, so the
// atomics are negligible; this is a pure HBM streaming read. 4 independent
// non-temporal b128 loads are kept in flight per thread.
__global__ void degree_accum(const vf4* __restrict__ R, const float* __restrict__ ew,
                             float* __restrict__ deg, int total4) {
    int i      = blockIdx.x * blockDim.x + threadIdx.x;
    int stride = gridDim.x * blockDim.x;
    for (; i + 3 * stride < total4; i += 4 * stride) {
        vf4 v0 = __builtin_nontemporal_load(&R[i]);
        vf4 v1 = __builtin_nontemporal_load(&R[i + stride]);
        vf4 v2 = __builtin_nontemporal_load(&R[i + 2 * stride]);
        vf4 v3 = __builtin_nontemporal_load(&R[i + 3 * stride]);
        deg_update(v0, i,              ew, deg);
        deg_update(v1, i + stride,     ew, deg);
        deg_update(v2, i + 2 * stride, ew, deg);
        deg_update(v3, i + 3 * stride, ew, deg);
    }
    for (; i < total4; i += stride) {
        vf4 v = __builtin_nontemporal_load(&R[i]);
        deg_update(v, i, ew, deg);
    }
}

// ---------------- fused MLP with fp32 WMMA ----------------
// One wave per 16-node tile. M[j][c] = (c<16 ? di : do)[row_j] * X[row_j][c&15].
// h = tanh(M @ W1 + b1) ; out = tanh(h @ W2 + b2).
// All register-array indices are compile-time constants; the lane-half (h)
// dependence lives only in load *addresses*.

#define W1_STRIDE 34   // 32 + 2 pad: conflict-free b64 lane pattern, 8B aligned
#define W2_STRIDE 66   // 64 + 2 pad
#define H_STRIDE  66   // 64 + 2 pad

__global__ __launch_bounds__(256) void mlp_wmma(
        const float* __restrict__ X,
        const float* __restrict__ degI, const float* __restrict__ degO,
        const float* __restrict__ W1, const float* __restrict__ B1,
        const float* __restrict__ W2, const float* __restrict__ B2,
        float* __restrict__ out) {
    __shared__ float sW1t[64 * W1_STRIDE];   // transposed: [col][k]
    __shared__ float sW2t[64 * W2_STRIDE];   // transposed: [col][k]
    __shared__ float sb1[64];
    __shared__ float sb2[64];
    __shared__ float sH[8][16 * H_STRIDE];   // per-wave h tile [row][col]

    const int tid = threadIdx.x;
    for (int i = tid; i < 32 * 64; i += 256) {
        int k = i >> 6, c = i & 63;
        sW1t[c * W1_STRIDE + k] = W1[i];
    }
    for (int i = tid; i < 64 * 64; i += 256) {
        int k = i >> 6, c = i & 63;
        sW2t[c * W2_STRIDE + k] = W2[i];
    }
    if (tid < 64) { sb1[tid] = B1[tid]; sb2[tid] = B2[tid]; }
    __syncthreads();

    const int wave = tid >> 5;
    const int lane = tid & 31;
    const int j    = lane & 15;   // row (A) / column (B,C,D) within tile
    const int h    = lane >> 4;   // lane half
    const int tile = blockIdx.x * 8 + wave;
    const int rowj = tile * 16 + j;

    // This lane only ever needs X columns {4t+2h, 4t+2h+1}: load exactly those.
    const float* xb = X + rowj * 16 + 2 * h;
    v2f x0 = *(const v2f*)(xb + 0);
    v2f x1 = *(const v2f*)(xb + 4);
    v2f x2 = *(const v2f*)(xb + 8);
    v2f x3 = *(const v2f*)(xb + 12);
    const float dv = degI[rowj];
    const float ov = degO[rowj];

    // A-fragments for layer 1 (K-step k uses column 4k+2h of M):
    // k=0..3 -> di * X-pair t=k ; k=4..7 -> do * X-pair t=k-4. All static indices.
    v2f ap[8];
    ap[0] = dv * x0;  ap[1] = dv * x1;  ap[2] = dv * x2;  ap[3] = dv * x3;
    ap[4] = ov * x0;  ap[5] = ov * x1;  ap[6] = ov * x2;  ap[7] = ov * x3;

    float* hrow = &sH[wave][0];

    // ---- layer 1: [16x32] @ [32x64], 4 col-tiles x 8 K-steps ----
#pragma unroll
    for (int nt = 0; nt < 4; ++nt) {
        const int col = nt * 16 + j;
        const float bias = sb1[col];
        v8f acc = { bias, bias, bias, bias, bias, bias, bias, bias };
        const v2f* wp = (const v2f*)(sW1t + col * W1_STRIDE + 2 * h);
#pragma unroll
        for (int k = 0; k < 8; ++k) {
            acc = __builtin_amdgcn_wmma_f32_16x16x4_f32(
                      false, ap[k], false, wp[2 * k], (short)0, acc, false, false);
        }
#pragma unroll
        for (int v = 0; v < 8; ++v)
            hrow[(v + 8 * h) * H_STRIDE + col] = tanhf(acc[v]);
    }
    __syncthreads();   // make h visible for the cross-lane A-fragment re-layout

    // ---- layer 2: [16x64] @ [64x64], 4 col-tiles x 16 K-steps ----
    const v2f* hp = (const v2f*)(hrow + j * H_STRIDE + 2 * h);
#pragma unroll
    for (int nt = 0; nt < 4; ++nt) {
        const int col = nt * 16 + j;
        const float bias = sb2[col];
        v8f acc = { bias, bias, bias, bias, bias, bias, bias, bias };
        const v2f* wp = (const v2f*)(sW2t + col * W2_STRIDE + 2 * h);
#pragma unroll
        for (int k = 0; k < 16; ++k) {
            acc = __builtin_amdgcn_wmma_f32_16x16x4_f32(
                      false, hp[2 * k], false, wp[2 * k], (short)0, acc, false, false);
        }
#pragma unroll
        for (int v = 0; v < 8; ++v)
            out[(tile * 16 + v + 8 * h) * 64 + col] = tanhf(acc[v]);
    }
}

// ---------------- launcher ----------------

extern "C" void kernel_launch(void* const* d_in, const int* in_sizes, int n_in,
                              void* d_out, int out_size, void* d_ws, size_t ws_size,
                              hipStream_t stream) {
    const float* X  = (const float*)d_in[0];
    const float* Ri = (const float*)d_in[1];
    const float* Ro = (const float*)d_in[2];
    const float* ew = (const float*)d_in[3];   // [B,E,1] -> flat [B*E]
    const float* W1 = (const float*)d_in[4];
    const float* B1 = (const float*)d_in[5];
    const float* W2 = (const float*)d_in[6];
    const float* B2 = (const float*)d_in[7];
    float* out  = (float*)d_out;

    float* degI = (float*)d_ws;          // [B*N]
    float* degO = degI + Bv * Nv;        // [B*N]

    const int ndeg = 2 * Bv * Nv;
    zero_ws<<<(ndeg + 255) / 256, 256, 0, stream>>>(degI, ndeg);

    const int total4 = (Bv * Ev * Nv) / 4;   // 16,777,216 float4 per matrix
    degree_accum<<<4096, 256, 0, stream>>>((const vf4*)Ri, ew, degI, total4);
    degree_accum<<<4096, 256, 0, stream>>>((const vf4*)Ro, ew, degO, total4);

    // 8192 rows -> 512 tiles of 16 -> 64 blocks x 8 waves
    mlp_wmma<<<64, 256, 0, stream>>>(X, degI, degO, W1, B1, W2, B2, out);
}